// SeqAttention_58686433133146
// MI455X (gfx1250) — compile-verified
//
#include <hip/hip_runtime.h>
#include <hip/hip_bf16.h>
#include <math.h>

typedef __attribute__((ext_vector_type(2))) float v2f;
typedef __attribute__((ext_vector_type(8))) float v8f;

namespace {

constexpr int kBH    = 128;
constexpr int kM     = 512;
constexpr int kD     = 128;
constexpr int kL     = 2048;
constexpr int kKLen  = kM + kL;   // 2560
constexpr int kWaves = 4;         // waves per workgroup, each owns 16 query rows
constexpr int kKPad  = 132;       // padded row (floats) for K/V LDS staging
constexpr float kScale = 0.088388347648318447f; // 1/sqrt(128)

__device__ __forceinline__ v8f wmma_f32(v2f a, v2f b, v8f c) {
  // D = A(16x4) * B(4x16) + C(16x16), fp32, wave32
  return __builtin_amdgcn_wmma_f32_16x16x4_f32(
      /*neg_a=*/false, a, /*neg_b=*/false, b,
      /*c_mod=*/(short)0, c, /*reuse_a=*/false, /*reuse_b=*/false);
}

} // namespace

__launch_bounds__(32 * kWaves)
__global__ void seq_attn_kernel(const float* __restrict__ Q,
                                const float* __restrict__ K,
                                const float* __restrict__ V,
                                const float* __restrict__ PE,
                                float* __restrict__ O)
{
  __shared__ __align__(16) float ldsK[kWaves][16][kKPad];
  __shared__ __align__(16) float ldsV[kWaves][16][kKPad];
  __shared__ __align__(16) float ldsP[kWaves][2][16][32]; // P ring, slot stride 512 floats
  __shared__ float ldsA[kWaves][16];                      // per-row online-softmax alpha

  const int tid  = threadIdx.x;
  const int w    = tid >> 5;   // wave index in workgroup
  const int lane = tid & 31;
  const int hi   = lane >> 4;  // half-wave select
  const int ln   = lane & 15;  // position within half

  const int b  = blockIdx.x;                          // batch-head
  const int m0 = blockIdx.y * (16 * kWaves) + w * 16; // wave's first query row

  const float* qb = Q + (size_t)b * kM * kD;
  const float* kb = K + (size_t)b * kKLen * kD;
  const float* vb = V + (size_t)b * kKLen * kD;
  float*       ob = O + (size_t)b * kM * kD;

  float* ringW = &ldsP[w][0][0][0];

  // zero ring slot 1 ("previous" tile at it==0) so the loop needs no special case
#pragma unroll
  for (int i = 0; i < 16; ++i) ringW[512 + 32 * i + lane] = 0.f;

  // ---- loop-invariant per-lane ring indices for the AV A-fragments.
  // Component k = 2*t + vreg covers abs col l = 4*t + vreg + 2*hi of the abs tile;
  // it reads rel-P element [row=ln][ (l-ln) & 15 ] from the "current" slot when
  // l >= ln, else from the "previous" slot (rescaled by this iteration's alpha).
  // Slot parity is applied later with a single XOR of bit 9.
  int  pidx[8];
  bool pprev[8];
#pragma unroll
  for (int t = 0; t < 4; ++t) {
#pragma unroll
    for (int vreg = 0; vreg < 2; ++vreg) {
      const int  l = 4 * t + vreg + 2 * hi;
      const bool x = (l < ln);
      pidx[2 * t + vreg]  = (x ? 512 : 0) + ln * 32 + ((l - ln) & 15);
      pprev[2 * t + vreg] = x;
    }
  }
  int sidx[8]; // P-tile store indices (C layout row j+8*hi, col ln), slot 0 base
#pragma unroll
  for (int j = 0; j < 8; ++j) sidx[j] = (j + 8 * hi) * 32 + ln;

  // ---- Q A-fragments.  A 16x4 layout: lane(hi,ln) -> M=ln, K = vreg + 2*hi.
  v2f qa[32];
  {
    const float* qr = qb + (size_t)(m0 + ln) * kD + 2 * hi;
#pragma unroll
    for (int dc = 0; dc < 32; ++dc) {
      qa[dc].x = qr[4 * dc + 0];
      qa[dc].y = qr[4 * dc + 1];
    }
  }

  const v8f vzero = {0.f, 0.f, 0.f, 0.f, 0.f, 0.f, 0.f, 0.f};
  v8f o[8];
#pragma unroll
  for (int i = 0; i < 8; ++i) o[i] = vzero;
  float rmax[8], rsum[8];
#pragma unroll
  for (int j = 0; j < 8; ++j) { rmax[j] = -3.0e38f; rsum[j] = 0.f; }

  // coalesced global -> LDS staging of a 16 x 128 f32 chunk
  auto load_chunk = [&](const float* src, int kc, float (*dst)[kKPad]) {
#pragma unroll
    for (int r = 0; r < 16; ++r) {
      const float4 t = *(const float4*)(src + (size_t)(kc + r) * kD + lane * 4);
      *(float4*)(&dst[r][lane * 4]) = t;
    }
  };

  auto lds_fence = [&]() { asm volatile("s_wait_dscnt 0" ::: "memory"); };

  // S_abs(16x16) = Q(16x128) * Kchunk^T(128x16); two independent accumulators
  // (even/odd d-chunks) halve the dependent WMMA->WMMA chain on the XDL pipe.
  auto gemm_qk = [&]() -> v8f {
    v8f acc0 = vzero, acc1 = vzero;
#pragma unroll
    for (int dc = 0; dc < 32; dc += 2) {
      v2f bf0, bf1;  // B[K=dd][N=ln] = key[kc+ln][4*dc+dd], dd = vreg + 2*hi
      bf0.x = ldsK[w][ln][4 * dc + 2 * hi];
      bf0.y = ldsK[w][ln][4 * dc + 1 + 2 * hi];
      bf1.x = ldsK[w][ln][4 * dc + 4 + 2 * hi];
      bf1.y = ldsK[w][ln][4 * dc + 5 + 2 * hi];
      acc0 = wmma_f32(qa[dc],     bf0, acc0);
      acc1 = wmma_f32(qa[dc + 1], bf1, acc1);
    }
#pragma unroll
    for (int j = 0; j < 8; ++j) acc0[j] += acc1[j];
    return acc0;
  };

  // prologue: absolute content tile at kc = m0
  load_chunk(kb, m0, ldsK[w]);
  lds_fence();
  v8f Tprev = gemm_qk();
  v8f Tnew;

#pragma unroll 1
  for (int it = 0; it < kL / 16; ++it) {
    const int lc  = it * 16;
    const int par = (it & 1) << 9;  // ring-slot parity bit (slot stride 512)

    // stage next K chunk (kc = m0+lc+16) and this iteration's V chunk (kc = m0+lc)
    load_chunk(kb, m0 + lc + 16, ldsK[w]);
    load_chunk(vb, m0 + lc,      ldsV[w]);
    __builtin_prefetch(kb + (size_t)(m0 + lc + 32) * kD + lane * 4, 0, 1);
    __builtin_prefetch(vb + (size_t)(m0 + lc + 16) * kD + lane * 4, 0, 1);

    // PE GEMM in relative coords (PE is 1MB -> L2 resident; reads coalesced),
    // also with a split accumulation chain.
    v8f spe;
    {
      v8f s0 = vzero, s1 = vzero;
#pragma unroll
      for (int dc = 0; dc < 32; dc += 2) {
        v2f bf0, bf1;  // B[K=dd][N=ln] = pe[4*dc+dd][lc+ln]
        bf0.x = PE[(size_t)(4 * dc + 2 * hi)     * kL + lc + ln];
        bf0.y = PE[(size_t)(4 * dc + 1 + 2 * hi) * kL + lc + ln];
        bf1.x = PE[(size_t)(4 * dc + 4 + 2 * hi) * kL + lc + ln];
        bf1.y = PE[(size_t)(4 * dc + 5 + 2 * hi) * kL + lc + ln];
        s0 = wmma_f32(qa[dc],     bf0, s0);
        s1 = wmma_f32(qa[dc + 1], bf1, s1);
      }
#pragma unroll
      for (int j = 0; j < 8; ++j) s0[j] += s1[j];
      spe = s0;
    }

    lds_fence();
    Tnew = gemm_qk();  // absolute tile at kc = m0 + lc + 16

    // assemble relative-coordinate scores: per C-layout row (m = m0+j+8*hi),
    // rel col l=lc+ln maps to abs col (ln + j + 8*hi) across Tprev/Tnew.
    float S[8];
#pragma unroll
    for (int j = 0; j < 8; ++j) {
      const int s   = j + 8 * hi;
      const int cc  = ln + s;
      const int src = hi * 16 + (cc & 15);
      const float a0 = __shfl(Tprev[j], src, 32);
      const float a1 = __shfl(Tnew[j],  src, 32);
      S[j] = ((cc < 16 ? a0 : a1) + spe[j]) * kScale;
    }

    // online softmax (reductions stay within 16-lane halves)
    float alpha[8];
#pragma unroll
    for (int j = 0; j < 8; ++j) {
      float mx = S[j];
      mx = fmaxf(mx, __shfl_xor(mx, 1, 32));
      mx = fmaxf(mx, __shfl_xor(mx, 2, 32));
      mx = fmaxf(mx, __shfl_xor(mx, 4, 32));
      mx = fmaxf(mx, __shfl_xor(mx, 8, 32));
      const float nm = fmaxf(rmax[j], mx);
      alpha[j] = __expf(rmax[j] - nm);
      rmax[j]  = nm;
      const float p = __expf(S[j] - nm);
      S[j] = p;
      float ps = p;
      ps += __shfl_xor(ps, 1, 32);
      ps += __shfl_xor(ps, 2, 32);
      ps += __shfl_xor(ps, 4, 32);
      ps += __shfl_xor(ps, 8, 32);
      rsum[j] = rsum[j] * alpha[j] + ps;
    }

#pragma unroll
    for (int dt = 0; dt < 8; ++dt)
#pragma unroll
      for (int j = 0; j < 8; ++j)
        o[dt][j] *= alpha[j];

    // publish alpha and the rel-P tile into ring slot `par`
    if (ln == 0) {
#pragma unroll
      for (int j = 0; j < 8; ++j) ldsA[w][8 * hi + j] = alpha[j];
    }
#pragma unroll
    for (int j = 0; j < 8; ++j) ringW[sidx[j] ^ par] = S[j];
    lds_fence();

    // AV GEMM on absolute tile kc = m0 + lc; skewed ring read, slot via XOR,
    // previous-tile elements rescaled by this iteration's alpha.
    const float aRow = ldsA[w][ln];
#pragma unroll
    for (int t = 0; t < 4; ++t) {
      v2f pa;
      pa.x = ringW[pidx[2 * t]     ^ par] * (pprev[2 * t]     ? aRow : 1.f);
      pa.y = ringW[pidx[2 * t + 1] ^ par] * (pprev[2 * t + 1] ? aRow : 1.f);
#pragma unroll
      for (int dt = 0; dt < 8; ++dt) {
        v2f bf;  // B[K=kk][N=ln] = V[kc + 4t + kk][dt*16 + ln]
        bf.x = ldsV[w][4 * t + 2 * hi][dt * 16 + ln];
        bf.y = ldsV[w][4 * t + 1 + 2 * hi][dt * 16 + ln];
        o[dt] = wmma_f32(pa, bf, o[dt]);
      }
    }

    Tprev = Tnew;
  }

  // epilogue: absolute tile kc = m0 + kL.  The last rel tile (it = 127) lives in
  // slot 1; with par = 0 the precomputed "previous" indices point exactly there.
  load_chunk(vb, m0 + kL, ldsV[w]);
  lds_fence();
#pragma unroll
  for (int t = 0; t < 4; ++t) {
    v2f pa;
    pa.x = pprev[2 * t]     ? ringW[pidx[2 * t]]     : 0.f;
    pa.y = pprev[2 * t + 1] ? ringW[pidx[2 * t + 1]] : 0.f;
#pragma unroll
    for (int dt = 0; dt < 8; ++dt) {
      v2f bf;
      bf.x = ldsV[w][4 * t + 2 * hi][dt * 16 + ln];
      bf.y = ldsV[w][4 * t + 1 + 2 * hi][dt * 16 + ln];
      o[dt] = wmma_f32(pa, bf, o[dt]);
    }
  }

  // normalize and store (C layout: row m0+j+8*hi, col dt*16+ln)
  float inv[8];
#pragma unroll
  for (int j = 0; j < 8; ++j) inv[j] = 1.0f / rsum[j];
#pragma unroll
  for (int dt = 0; dt < 8; ++dt) {
#pragma unroll
    for (int j = 0; j < 8; ++j) {
      ob[(size_t)(m0 + j + 8 * hi) * kD + dt * 16 + ln] = o[dt][j] * inv[j];
    }
  }
}

extern "C" void kernel_launch(void* const* d_in, const int* in_sizes, int n_in,
                              void* d_out, int out_size, void* d_ws, size_t ws_size,
                              hipStream_t stream) {
  (void)in_sizes; (void)n_in; (void)d_ws; (void)ws_size; (void)out_size;
  const float* q  = (const float*)d_in[0];
  const float* k  = (const float*)d_in[1];
  const float* v  = (const float*)d_in[2];
  const float* pe = (const float*)d_in[3];
  float* out = (float*)d_out;
  dim3 grid(kBH, kM / (16 * kWaves));
  dim3 block(32 * kWaves);
  seq_attn_kernel<<<grid, block, 0, stream>>>(q, k, v, pe, out);
}